// HouseholderLayer_73847667687731
// MI455X (gfx1250) — compile-verified
//
#include <hip/hip_runtime.h>

typedef __attribute__((ext_vector_type(2))) float v2f;
typedef __attribute__((ext_vector_type(8))) float v8f;

#define CIN   2048
#define VDIM  128
#define NROWS 16384

// ---------------------------------------------------------------------------
// Kernel 1: normalize each column of Fq -> V (2048x128 row-major) and
//           Vt (128x2048 row-major, the transpose).
// ---------------------------------------------------------------------------
__global__ __launch_bounds__(256) void hh_normalize(const float* __restrict__ Fq,
                                                    float* __restrict__ V,
                                                    float* __restrict__ Vt) {
    __shared__ float red[256];
    const int j   = blockIdx.x;      // column 0..127
    const int tid = threadIdx.x;

    float s = 0.0f;
    for (int i = tid; i < CIN; i += 256) {
        float v = Fq[i * VDIM + j];
        s += v * v;
    }
    red[tid] = s;
    __syncthreads();
    for (int off = 128; off > 0; off >>= 1) {
        if (tid < off) red[tid] += red[tid + off];
        __syncthreads();
    }
    const float inv = 1.0f / sqrtf(red[0]);
    for (int i = tid; i < CIN; i += 256) {
        float v = Fq[i * VDIM + j] * inv;
        V[i * VDIM + j]  = v;
        Vt[j * CIN + i]  = v;
    }
}

// ---------------------------------------------------------------------------
// Kernel 2: Gram matrix S = Vt * V  (S[a][b] = v_a . v_b), 128x128.
// ---------------------------------------------------------------------------
__global__ __launch_bounds__(64) void hh_gram(const float* __restrict__ Vt,
                                              float* __restrict__ S) {
    __shared__ float red[64];
    const int a = blockIdx.x, b = blockIdx.y, tid = threadIdx.x;
    const float* pa = Vt + a * CIN;
    const float* pb = Vt + b * CIN;
    float s = 0.0f;
    for (int i = tid; i < CIN; i += 64) s += pa[i] * pb[i];
    red[tid] = s;
    __syncthreads();
    for (int off = 32; off > 0; off >>= 1) {
        if (tid < off) red[tid] += red[tid + off];
        __syncthreads();
    }
    if (tid == 0) S[a * VDIM + b] = red[0];
}

// ---------------------------------------------------------------------------
// Kernel 3: build upper-triangular T (compact-WY, tau = 2 for all columns):
//   T[k][k] = 2 ;  T[0:k,k] = -2 * T[0:k,0:k] @ S[0:k,k]
// Single workgroup, 128 threads, T kept in LDS.
// ---------------------------------------------------------------------------
__global__ __launch_bounds__(128) void hh_tmat(const float* __restrict__ S,
                                               float* __restrict__ T) {
    __shared__ float Ts[VDIM * VDIM];   // 64 KB
    const int tid = threadIdx.x;        // row index 0..127
    for (int k = 0; k < VDIM; ++k) Ts[tid * VDIM + k] = 0.0f;
    __syncthreads();
    if (tid == 0) Ts[0] = 2.0f;
    __syncthreads();
    for (int k = 1; k < VDIM; ++k) {
        if (tid < k) {
            float t = 0.0f;
            for (int j = tid; j < k; ++j)
                t += Ts[tid * VDIM + j] * S[j * VDIM + k];
            Ts[tid * VDIM + k] = -2.0f * t;
        } else if (tid == k) {
            Ts[k * VDIM + k] = 2.0f;
        }
        __syncthreads();
    }
    for (int k = 0; k < VDIM; ++k) T[tid * VDIM + k] = Ts[tid * VDIM + k];
}

// ---------------------------------------------------------------------------
// Kernel 4 (fused main pass):  out = x - ((x V) T) V^T + by
// One wave owns a 16-row slab. All GEMMs use V_WMMA_F32_16X16X4_F32.
//  Phase A: P(16x128)  = x_slab(16x2048) @ V(2048x128)    -> stage P in LDS
//  Phase B: W(16x128)  = P @ T(128x128)                    -> stage W in LDS
//  Phase C: out_slab   = x_slab - W @ Vt(128x2048) + by
// ---------------------------------------------------------------------------
__global__ __launch_bounds__(256) void hh_apply(const float* __restrict__ x,
                                                const float* __restrict__ V,
                                                const float* __restrict__ Vt,
                                                const float* __restrict__ T,
                                                const float* __restrict__ by,
                                                float* __restrict__ out) {
    __shared__ float stage[8 * 16 * VDIM];          // 8 waves * 8 KB = 64 KB
    const int lane    = threadIdx.x & 31;
    const int wave    = threadIdx.x >> 5;
    const int lh      = lane >> 4;                  // half-wave: 0 or 1
    const int ln      = lane & 15;                  // lane-in-half: 0..15
    const int rowbase = (blockIdx.x * 8 + wave) * 16;
    float* P = stage + wave * (16 * VDIM);          // per-wave staging tile

    const v8f vzero = {0.f, 0.f, 0.f, 0.f, 0.f, 0.f, 0.f, 0.f};

    // ---------------- Phase A: P = x_slab @ V -------------------------------
    v8f acc[8];
#pragma unroll
    for (int jt = 0; jt < 8; ++jt) acc[jt] = vzero;

    const float* ax = x + (size_t)(rowbase + ln) * CIN + 2 * lh;  // A row ptr
    for (int k0 = 0; k0 < CIN; k0 += 4) {
        v2f a = *(const v2f*)(ax + k0);
#pragma unroll
        for (int jt = 0; jt < 8; ++jt) {
            // B[k][n] = V[k0+k][jt*16+n] = Vt[(jt*16+n)*CIN + k0+k]
            v2f b = *(const v2f*)(Vt + (jt * 16 + ln) * CIN + 2 * lh + k0);
            acc[jt] = __builtin_amdgcn_wmma_f32_16x16x4_f32(
                false, a, false, b, (short)0, acc[jt], false, false);
        }
    }
    // C-layout -> row-major LDS:  element (m = g + 8*lh, n = ln) of tile jt
#pragma unroll
    for (int jt = 0; jt < 8; ++jt)
#pragma unroll
        for (int g = 0; g < 8; ++g)
            P[(g + 8 * lh) * VDIM + jt * 16 + ln] = acc[jt][g];

    // ---------------- Phase B: W = P @ T ------------------------------------
    v8f w[8];
#pragma unroll
    for (int kt = 0; kt < 8; ++kt) w[kt] = vzero;

    for (int j0 = 0; j0 < VDIM; j0 += 4) {
        v2f a = *(const v2f*)(P + ln * VDIM + j0 + 2 * lh);     // ds_load_b64
#pragma unroll
        for (int kt = 0; kt < 8; ++kt) {
            v2f b;
            b.x = T[(j0 + 2 * lh)     * VDIM + kt * 16 + ln];
            b.y = T[(j0 + 2 * lh + 1) * VDIM + kt * 16 + ln];
            w[kt] = __builtin_amdgcn_wmma_f32_16x16x4_f32(
                false, a, false, b, (short)0, w[kt], false, false);
        }
    }
    // overwrite stage with W (all P reads are program-order before these)
#pragma unroll
    for (int kt = 0; kt < 8; ++kt)
#pragma unroll
        for (int g = 0; g < 8; ++g)
            P[(g + 8 * lh) * VDIM + kt * 16 + ln] = w[kt][g];

    // ---------------- Phase C: out = x - W @ V^T + by -----------------------
    for (int cb = 0; cb < CIN / 16; ++cb) {
        const int cbase = cb * 16;
        v8f c = vzero;
        for (int k0 = 0; k0 < VDIM; k0 += 4) {
            v2f a = *(const v2f*)(P + ln * VDIM + k0 + 2 * lh); // W frag (LDS)
            // B[k][n] = Vt[k0+k][cbase+n] = V[(cbase+n)*VDIM + k0+k]
            v2f b = *(const v2f*)(V + (cbase + ln) * VDIM + k0 + 2 * lh);
            c = __builtin_amdgcn_wmma_f32_16x16x4_f32(
                false, a, false, b, (short)0, c, false, false);
        }
        const float bias = by[cbase + ln];
#pragma unroll
        for (int g = 0; g < 8; ++g) {
            const size_t idx = (size_t)(rowbase + g + 8 * lh) * CIN + cbase + ln;
            out[idx] = x[idx] - c[g] + bias;
        }
    }
}

// ---------------------------------------------------------------------------
extern "C" void kernel_launch(void* const* d_in, const int* in_sizes, int n_in,
                              void* d_out, int out_size, void* d_ws, size_t ws_size,
                              hipStream_t stream) {
    const float* x  = (const float*)d_in[0];   // [16384, 2048]
    const float* Fq = (const float*)d_in[1];   // [2048, 128]
    const float* by = (const float*)d_in[2];   // [2048]
    float* out = (float*)d_out;

    float* V  = (float*)d_ws;                  // 2048*128
    float* Vt = V  + (size_t)CIN * VDIM;       // 128*2048
    float* S  = Vt + (size_t)CIN * VDIM;       // 128*128
    float* T  = S  + (size_t)VDIM * VDIM;      // 128*128
    // total workspace: ~2.2 MB

    hh_normalize<<<VDIM, 256, 0, stream>>>(Fq, V, Vt);
    hh_gram<<<dim3(VDIM, VDIM), 64, 0, stream>>>(Vt, S);
    hh_tmat<<<1, VDIM, 0, stream>>>(S, T);
    hh_apply<<<NROWS / 128, 256, 0, stream>>>(x, V, Vt, T, by, out);
}